// tableInterpolation_27968827031873
// MI455X (gfx1250) — compile-verified
//
#include <hip/hip_runtime.h>

// Bilinear "table interpolation" with y_scale == 0 (faithful to reference):
//   qy = 0 * (...)  -> fy = 0, ay = 0, iy = 0 for all finite inputs
//   => out[i] = g[0][ix] + (g[0][ix+1] - g[0][ix]) * ax,   ix in {0,1}
// The kernel streams only v_query (4 B/elem) and writes out (4 B/elem):
// 64 MB total => ~2.75 us at 23.3 TB/s HBM. Pure bandwidth-bound.
//
// MLP structure: each thread owns ITER=8 float4 packets (stride = total
// threads, so every global_load_b128 is unit-dense across the wave). All 8
// loads are issued back-to-back (128 B in flight per thread, 4 KB per wave)
// before any compute, then 8 global_store_b128 fire-and-forget. Grid values
// live in registers; the grid "gather" is two v_cndmask selects per element.
// Default RT temporal hints: the 64 MB working set stays resident in the
// 192 MB L2 across graph replays.

__device__ __forceinline__ float interp1(float v, float b10, float invx,
                                         float g00, float g01, float g02) {
  float qx = (v - b10) * invx;                 // x_scale folded into invx
  float fx = floorf(qx);
  fx = fminf(fmaxf(fx, 0.0f), 1.0f);           // clip(floor(qx), 0, W-2=1)
  float ax = fminf(fmaxf(qx - fx, 0.0f), 1.0f);
  bool hi = (fx >= 1.0f);                      // ix in {0,1}
  float tl = hi ? g01 : g00;
  float tr = hi ? g02 : g01;
  return __builtin_fmaf(ax, tr - tl, tl);      // tl + (tr-tl)*ax
}

__device__ __forceinline__ float4 interp4(float4 v, float b10, float invx,
                                          float g00, float g01, float g02) {
  float4 r;
  r.x = interp1(v.x, b10, invx, g00, g01, g02);
  r.y = interp1(v.y, b10, invx, g00, g01, g02);
  r.z = interp1(v.z, b10, invx, g00, g01, g02);
  r.w = interp1(v.w, b10, invx, g00, g01, g02);
  return r;
}

#define ITER 8

__global__ __launch_bounds__(256) void tableInterp_kernel(
    const float* __restrict__ vq,
    const float* __restrict__ grid,
    const float* __restrict__ bounds,
    float* __restrict__ out,
    int n) {
  // Uniform scalars (s_load; amortized over 32 elements/thread).
  const float b10 = bounds[2];
  const float b11 = bounds[3];
  const float invx = 2.0f / (b11 - b10);       // x_scale = GRID_W - 1 = 2
  const float g00 = grid[0], g01 = grid[1], g02 = grid[2];

  const int n4 = n >> 2;                       // number of float4 packets
  const int T = gridDim.x * blockDim.x;
  const int t0 = blockIdx.x * blockDim.x + threadIdx.x;

  const float4* __restrict__ v4 = reinterpret_cast<const float4*>(vq);
  float4* __restrict__ o4 = reinterpret_cast<float4*>(out);

  if (t0 + (ITER - 1) * T < n4) {
    // Fast path (taken by all threads for N = 8,388,608, where n4 == ITER*T):
    // batch all loads first for 8 outstanding global_load_b128 per thread.
    float4 v[ITER];
#pragma unroll
    for (int k = 0; k < ITER; ++k) v[k] = v4[t0 + k * T];
#pragma unroll
    for (int k = 0; k < ITER; ++k)
      o4[t0 + k * T] = interp4(v[k], b10, invx, g00, g01, g02);
  } else {
    for (int i = t0; i < n4; i += T)
      o4[i] = interp4(v4[i], b10, invx, g00, g01, g02);
  }

  // Tail (n not multiple of 4) — not taken for N = 8,388,608.
  const int rem = n & 3;
  if (t0 < rem) {
    const int i = n4 * 4 + t0;
    out[i] = interp1(vq[i], b10, invx, g00, g01, g02);
  }
}

extern "C" void kernel_launch(void* const* d_in, const int* in_sizes, int n_in,
                              void* d_out, int out_size, void* d_ws, size_t ws_size,
                              hipStream_t stream) {
  // Inputs (setup_inputs order): 0=x_query (unused: y_scale==0 makes the
  // output exactly independent of it), 1=v_query, 2=grid, 3=bounds.
  const float* vq     = (const float*)d_in[1];
  const float* grid   = (const float*)d_in[2];
  const float* bounds = (const float*)d_in[3];
  float* out = (float*)d_out;

  const int n = in_sizes[1];                   // 8,388,608
  const long long n4 = (long long)n >> 2;      // 2,097,152 float4 packets

  const int block = 256;
  long long want_threads = (n4 + ITER - 1) / ITER;   // exactly n4/8 here
  int blocks = (int)((want_threads + block - 1) / block);
  if (blocks < 1) blocks = 1;

  tableInterp_kernel<<<blocks, block, 0, stream>>>(vq, grid, bounds, out, n);
}